// SpatialCGNL_1133871366206
// MI455X (gfx1250) — compile-verified
//
#include <hip/hip_runtime.h>

// SpatialCGNL fused pipeline for MI455X (gfx1250, wave32, WMMA).
// Shapes: B=16, CIN=512, H=W=64 (HW=4096), CP=256, G=8.
//   k1: t/p/g = W{t,p,g} @ x  (3x (256x512)x(512x4096) GEMM per batch, f32 WMMA,
//       register-blocked 2 M-tiles x 2 N-tiles per wave -> 12 WMMA per K-step)
//       + att[b,g] = sum(p_g * g_g)   (wave reduce + atomic f32)
//   k2: z = att * (Wz_grouped @ t)    (8x (64x32)x(32x4096) GEMM per batch)
//       + per-group sum / sumsq for GroupNorm
//   k3: out = (z - mu) * rsqrt(var+eps) * gamma + beta + x

typedef float v2f __attribute__((ext_vector_type(2)));
typedef float v8f __attribute__((ext_vector_type(8)));

#define NB   16
#define CIN  512
#define CP   256
#define HW   4096
#define NG   8

#define WMMA_F32(A, Bf, C) \
  __builtin_amdgcn_wmma_f32_16x16x4_f32(false, (A), false, (Bf), (short)0, (C), false, false)

__global__ __launch_bounds__(256) void k0_zero(float* __restrict__ p, int n) {
  int i = blockIdx.x * blockDim.x + threadIdx.x;
  if (i < n) p[i] = 0.0f;
}

// grid (HW/32=128, NB), block 256 (8 waves). Wave w owns CP rows [32w,32w+32)
// == attention group w, and a 32-column spatial slab shared via LDS.
__global__ __launch_bounds__(256) void k1_tpg_att(
    const float* __restrict__ x,  const float* __restrict__ Wt,
    const float* __restrict__ Wp, const float* __restrict__ Wg,
    float* __restrict__ t_ws, float* __restrict__ att)
{
  __shared__ float sB[32 * 33];   // x chunk: 32 K-rows x 32 cols, +1 pad vs banks

  const int b    = blockIdx.y;
  const int n0   = blockIdx.x * 32;
  const int tid  = threadIdx.x;
  const int wave = tid >> 5;
  const int lane = tid & 31;
  const int col  = lane & 15;           // A row-in-tile / B column / C column
  const int kh   = (lane >> 4) << 1;    // K sub-offset 0 or 2 (f32 WMMA layout)
  const int m0   = wave * 32;           // CP row base (== group w * 32)

  const float* xb = x + ((size_t)b * CIN) * HW + n0;

  v8f ct00 = {}, ct01 = {}, ct10 = {}, ct11 = {};
  v8f cp00 = {}, cp01 = {}, cp10 = {}, cp11 = {};
  v8f cg00 = {}, cg01 = {}, cg10 = {}, cg11 = {};

  for (int kc = 0; kc < CIN; kc += 32) {
    // cooperative stage of X[kc..kc+32) x [n0..n0+32): 1024 floats, 4/thread
#pragma unroll
    for (int j = 0; j < 4; ++j) {
      const int i = tid + j * 256;
      sB[(i >> 5) * 33 + (i & 31)] = xb[(size_t)(kc + (i >> 5)) * HW + (i & 31)];
    }
    if (kc + 32 < CIN)  // global_prefetch_b8 of next chunk's cachelines
      __builtin_prefetch(&xb[(size_t)(kc + 32 + (tid >> 3)) * HW + ((tid & 7) << 2)], 0, 1);
    __syncthreads();

#pragma unroll
    for (int kk = 0; kk < 8; ++kk) {
      const int kl = kk * 4 + kh;       // K within chunk for this lane half
      v2f b0, b1;
      b0.x = sB[kl * 33 + col];        b0.y = sB[(kl + 1) * 33 + col];
      b1.x = sB[kl * 33 + 16 + col];   b1.y = sB[(kl + 1) * 33 + 16 + col];

      const int kg = kc + kl;           // global K (even -> 8B aligned v2f)
      const int r0 = m0 + col;
      const int r1 = m0 + 16 + col;
      v2f at0 = *(const v2f*)&Wt[(size_t)r0 * CIN + kg];
      v2f at1 = *(const v2f*)&Wt[(size_t)r1 * CIN + kg];
      v2f ap0 = *(const v2f*)&Wp[(size_t)r0 * CIN + kg];
      v2f ap1 = *(const v2f*)&Wp[(size_t)r1 * CIN + kg];
      v2f ag0 = *(const v2f*)&Wg[(size_t)r0 * CIN + kg];
      v2f ag1 = *(const v2f*)&Wg[(size_t)r1 * CIN + kg];

      ct00 = WMMA_F32(at0, b0, ct00);  ct01 = WMMA_F32(at0, b1, ct01);
      ct10 = WMMA_F32(at1, b0, ct10);  ct11 = WMMA_F32(at1, b1, ct11);
      cp00 = WMMA_F32(ap0, b0, cp00);  cp01 = WMMA_F32(ap0, b1, cp01);
      cp10 = WMMA_F32(ap1, b0, cp10);  cp11 = WMMA_F32(ap1, b1, cp11);
      cg00 = WMMA_F32(ag0, b0, cg00);  cg01 = WMMA_F32(ag0, b1, cg01);
      cg10 = WMMA_F32(ag1, b0, cg10);  cg11 = WMMA_F32(ag1, b1, cg11);
    }
    __syncthreads();
  }

  // store t tiles. C/D layout: VGPR v: lanes 0-15 -> M=v, lanes 16-31 -> M=v+8
  float* tb = t_ws + ((size_t)b * CP) * HW + n0;
  const int rofs = (lane >> 4) << 3;
#pragma unroll
  for (int v = 0; v < 8; ++v) {
    tb[(size_t)(m0 + v + rofs) * HW + col]           = ct00[v];
    tb[(size_t)(m0 + v + rofs) * HW + 16 + col]      = ct01[v];
    tb[(size_t)(m0 + 16 + v + rofs) * HW + col]      = ct10[v];
    tb[(size_t)(m0 + 16 + v + rofs) * HW + 16 + col] = ct11[v];
  }

  // att[b, wave] += sum(p * g) over this 32x32 tile
  float s = 0.0f;
#pragma unroll
  for (int v = 0; v < 8; ++v)
    s += cp00[v] * cg00[v] + cp01[v] * cg01[v] + cp10[v] * cg10[v] + cp11[v] * cg11[v];
#pragma unroll
  for (int m = 16; m >= 1; m >>= 1) s += __shfl_xor(s, m, 32);
  if (lane == 0) atomicAdd(&att[b * NG + wave], s);
}

// grid (HW/256=16, G=8, B=16), block 256 (8 waves). Wave w: 32 cols,
// all 64 output channels of group g (4 M-tiles x 2 N-tiles), K=32.
__global__ __launch_bounds__(256) void k2_z_stats(
    const float* __restrict__ t_ws, const float* __restrict__ Wz,
    const float* __restrict__ att,  float* __restrict__ z_ws,
    float* __restrict__ gsum, float* __restrict__ gsq)
{
  const int b    = blockIdx.z;
  const int g    = blockIdx.y;
  const int tid  = threadIdx.x;
  const int wave = tid >> 5;
  const int lane = tid & 31;
  const int col  = lane & 15;
  const int kh   = (lane >> 4) << 1;
  const int n0   = blockIdx.x * 256 + wave * 32;

  const float* tg = t_ws + ((size_t)b * CP + g * 32) * HW + n0;
  const float* wz = Wz + (size_t)g * 64 * 32;   // (64 out, 32 in) row-major

  v8f c0a = {}, c1a = {}, c2a = {}, c3a = {};
  v8f c0b = {}, c1b = {}, c2b = {}, c3b = {};

#pragma unroll
  for (int kk = 0; kk < 8; ++kk) {
    const int kl = kk * 4 + kh;                 // even -> aligned v2f
    v2f b0, b1;
    b0.x = tg[(size_t)kl * HW + col];
    b0.y = tg[(size_t)(kl + 1) * HW + col];
    b1.x = tg[(size_t)kl * HW + 16 + col];
    b1.y = tg[(size_t)(kl + 1) * HW + 16 + col];
    v2f a0 = *(const v2f*)&wz[(size_t)(col) * 32 + kl];
    v2f a1 = *(const v2f*)&wz[(size_t)(16 + col) * 32 + kl];
    v2f a2 = *(const v2f*)&wz[(size_t)(32 + col) * 32 + kl];
    v2f a3 = *(const v2f*)&wz[(size_t)(48 + col) * 32 + kl];
    c0a = WMMA_F32(a0, b0, c0a);  c0b = WMMA_F32(a0, b1, c0b);
    c1a = WMMA_F32(a1, b0, c1a);  c1b = WMMA_F32(a1, b1, c1b);
    c2a = WMMA_F32(a2, b0, c2a);  c2b = WMMA_F32(a2, b1, c2b);
    c3a = WMMA_F32(a3, b0, c3a);  c3b = WMMA_F32(a3, b1, c3b);
  }

  const float scale = att[b * NG + g];          // z = att * (Wz @ t)
  float* zb = z_ws + ((size_t)b * CIN + g * 64) * HW + n0;
  const int rofs = (lane >> 4) << 3;
  float s1 = 0.0f, s2 = 0.0f;
#pragma unroll
  for (int v = 0; v < 8; ++v) {
    float z0a = c0a[v] * scale, z1a = c1a[v] * scale;
    float z2a = c2a[v] * scale, z3a = c3a[v] * scale;
    float z0b = c0b[v] * scale, z1b = c1b[v] * scale;
    float z2b = c2b[v] * scale, z3b = c3b[v] * scale;
    s1 += z0a + z1a + z2a + z3a + z0b + z1b + z2b + z3b;
    s2 += z0a * z0a + z1a * z1a + z2a * z2a + z3a * z3a
        + z0b * z0b + z1b * z1b + z2b * z2b + z3b * z3b;
    zb[(size_t)(v + rofs) * HW + col]           = z0a;
    zb[(size_t)(16 + v + rofs) * HW + col]      = z1a;
    zb[(size_t)(32 + v + rofs) * HW + col]      = z2a;
    zb[(size_t)(48 + v + rofs) * HW + col]      = z3a;
    zb[(size_t)(v + rofs) * HW + 16 + col]      = z0b;
    zb[(size_t)(16 + v + rofs) * HW + 16 + col] = z1b;
    zb[(size_t)(32 + v + rofs) * HW + 16 + col] = z2b;
    zb[(size_t)(48 + v + rofs) * HW + 16 + col] = z3b;
  }
#pragma unroll
  for (int m = 16; m >= 1; m >>= 1) {
    s1 += __shfl_xor(s1, m, 32);
    s2 += __shfl_xor(s2, m, 32);
  }
  if (lane == 0) {
    atomicAdd(&gsum[b * NG + g], s1);
    atomicAdd(&gsq[b * NG + g], s2);
  }
}

// float4 elementwise: out = (z - mu)*rsqrt(var+eps)*gamma + beta + x
__global__ __launch_bounds__(256) void k3_norm(
    const float* __restrict__ z_ws, const float* __restrict__ x,
    const float* __restrict__ gsum, const float* __restrict__ gsq,
    const float* __restrict__ gamma, const float* __restrict__ beta,
    float* __restrict__ out)
{
  const size_t i4 = (size_t)blockIdx.x * blockDim.x + threadIdx.x;  // < 2^23
  const int c = (int)((i4 >> 10) & 511);
  const int b = (int)(i4 >> 19);
  const int grp = c >> 6;
  const float inv_cnt = 1.0f / (64.0f * 4096.0f);
  const float mu  = gsum[b * NG + grp] * inv_cnt;
  const float var = gsq[b * NG + grp] * inv_cnt - mu * mu;
  const float rs  = __frsqrt_rn(var + 1e-5f);
  const float ga  = gamma[c] * rs;
  const float be  = beta[c] - mu * ga;

  const float4 z  = ((const float4*)z_ws)[i4];
  const float4 xv = ((const float4*)x)[i4];
  float4 o;
  o.x = z.x * ga + be + xv.x;
  o.y = z.y * ga + be + xv.y;
  o.z = z.z * ga + be + xv.z;
  o.w = z.w * ga + be + xv.w;
  ((float4*)out)[i4] = o;
}

extern "C" void kernel_launch(void* const* d_in, const int* in_sizes, int n_in,
                              void* d_out, int out_size, void* d_ws, size_t ws_size,
                              hipStream_t stream) {
  const float* x     = (const float*)d_in[0];
  const float* Wt    = (const float*)d_in[1];
  const float* Wp    = (const float*)d_in[2];
  const float* Wg    = (const float*)d_in[3];
  const float* Wz    = (const float*)d_in[4];
  const float* gamma = (const float*)d_in[5];
  const float* beta  = (const float*)d_in[6];
  float* out = (float*)d_out;

  float* t_ws = (float*)d_ws;                          // 16*256*4096 f32 (64 MiB)
  float* z_ws = t_ws + (size_t)NB * CP * HW;           // 16*512*4096 f32 (128 MiB)
  float* att  = z_ws + (size_t)NB * CIN * HW;          // 128 f32
  float* gsum = att + 128;                             // 128 f32
  float* gsq  = gsum + 128;                            // 128 f32

  k0_zero<<<2, 256, 0, stream>>>(att, 384);
  k1_tpg_att<<<dim3(HW / 32, NB), 256, 0, stream>>>(x, Wt, Wp, Wg, t_ws, att);
  k2_z_stats<<<dim3(HW / 256, NG, NB), 256, 0, stream>>>(t_ws, Wz, att, z_ws, gsum, gsq);
  k3_norm<<<(NB * CIN * HW / 4) / 256, 256, 0, stream>>>(z_ws, x, gsum, gsq, gamma, beta, out);
}